// AMSoftmaxLoss_65068754535073
// MI455X (gfx1250) — compile-verified
//
#include <hip/hip_runtime.h>
#include <hip/hip_bf16.h>
#include <stdint.h>

// ---------------------------------------------------------------------------
// AM-Softmax loss, MI455X (gfx1250).
// Memory-bound streaming reduction: 412 MB @ 23.3 TB/s ~= 18 us floor.
// - CDNA5 async global->LDS copies (ASYNCcnt) hide HBM latency, double-buffered
//   per-wave so no cross-wave barriers are needed in the stream loop.
// - Whole stream is non-temporal: 412 MB touched once > 192 MB L2, no reuse.
// - exp(S*x) lowered to v_exp_f32 via exp2; 4-way split accumulators let the
//   full-rate adds pipeline behind the quarter-rate TRANS exps.
// ---------------------------------------------------------------------------

constexpr int   N_ROWS     = 2048;
constexpr int   C_COLS     = 50257;
constexpr int   ROW_BYTES  = C_COLS * 4;            // 201028 (only 4B aligned!)
constexpr int   THREADS    = 256;                   // 8 wave32s
constexpr int   WAVES      = THREADS / 32;
constexpr int   WAVE_TILE  = 512;                   // floats per wave per stage (2KB)
constexpr int   BLOCK_TILE = WAVES * WAVE_TILE;     // 4096 floats per stage

constexpr float S_SCALE = 30.0f;
constexpr float LOG2E   = 1.4426950408889634f;
constexpr float LN2     = 0.6931471805599453f;
constexpr float KS      = S_SCALE * LOG2E;          // exp(S*x) = exp2(KS*x)

__device__ __forceinline__ float ntload(const float* p) {
    return __builtin_nontemporal_load(p);
}

// Issue one stage: 4 x b128 async loads per lane -> 2KB contiguous into this
// wave's LDS buffer. INST_OFFSET advances BOTH the LDS and global addresses
// (CDNA5 ISA GVS pseudocode), so one (lds,voff) pair covers 2KB.
// s_wait_dscnt 0 first: guarantee prior ds_loads from this buffer retired
// before the async engine may overwrite it (WAR on LDS). NT temporal hint:
// stream-once data, bypass cache retention.
__device__ __forceinline__ void async_stage(uint32_t lds_addr, uint32_t goff,
                                            const float* base) {
    asm volatile(
        "s_wait_dscnt 0x0\n\t"
        "global_load_async_to_lds_b128 %0, %1, %2 th:TH_LOAD_NT\n\t"
        "global_load_async_to_lds_b128 %0, %1, %2 offset:512 th:TH_LOAD_NT\n\t"
        "global_load_async_to_lds_b128 %0, %1, %2 offset:1024 th:TH_LOAD_NT\n\t"
        "global_load_async_to_lds_b128 %0, %1, %2 offset:1536 th:TH_LOAD_NT"
        :
        : "v"(lds_addr), "v"(goff), "s"(base)
        : "memory");
}

__device__ __forceinline__ void wait_async_le4() {
    // 4 loads of the *next* stage may remain in flight; async loads complete
    // in order, so cnt<=4 implies the current stage's 4 loads have landed.
    asm volatile("s_wait_asynccnt 0x4" ::: "memory");
}
__device__ __forceinline__ void wait_async_le0() {
    asm volatile("s_wait_asynccnt 0x0" ::: "memory");
}

__global__ __launch_bounds__(THREADS)
void amsoftmax_row_kernel(const float* __restrict__ score,
                          const int*   __restrict__ labels,
                          float*       __restrict__ rowL) {
    __shared__ __align__(16) float stage[2][WAVES][WAVE_TILE]; // 32KB
    __shared__ float red[THREADS];

    const int row  = blockIdx.x;
    const int tid  = threadIdx.x;
    const int wave = tid >> 5;
    const int lane = tid & 31;

    const uint32_t rowbase = (uint32_t)row * (uint32_t)ROW_BYTES;

    // Peel 0..3 leading floats so the b128 async stages are 16B aligned.
    const int peel = (int)(((16u - (rowbase & 15u)) & 15u) >> 2);
    const int full_stages = (C_COLS - peel) / BLOCK_TILE;      // 12
    const int tail_start  = peel + full_stages * BLOCK_TILE;

    const size_t row_elem = (size_t)row * (size_t)C_COLS;

    float a0 = 0.0f, a1 = 0.0f, a2 = 0.0f, a3 = 0.0f;

    // prologue (at most 3 elements)
    for (int i = tid; i < peel; i += THREADS)
        a0 += __builtin_amdgcn_exp2f(KS * ntload(&score[row_elem + i]));

    // LDS byte addresses (wave-relative; hardware adds LDS_BASE): low 32 bits
    // of the flat shared address are the LDS offset.
    const uint32_t lds0 =
        (uint32_t)(uintptr_t)(&stage[0][wave][0]) + (uint32_t)(lane * 16);
    const uint32_t lds1 =
        (uint32_t)(uintptr_t)(&stage[1][wave][0]) + (uint32_t)(lane * 16);

    // 32-bit global byte offset of this lane's first b128 within stage 0.
    const uint32_t goff0 = rowbase + (uint32_t)(peel * 4)
                         + (uint32_t)(wave * WAVE_TILE * 4 + lane * 16);

    // Prime the pipeline: stage 0 -> buffer 0.
    async_stage(lds0, goff0, score);

    for (int s = 0; s < full_stages; ++s) {
        if (s + 1 < full_stages) {
            const uint32_t g = goff0 + (uint32_t)((s + 1) * BLOCK_TILE * 4);
            async_stage(((s + 1) & 1) ? lds1 : lds0, g, score);
            wait_async_le4();
        } else {
            wait_async_le0();
        }
        const float* buf = &stage[s & 1][wave][0];
        // lane-strided: 32 consecutive dwords per ds access -> conflict-free.
        // 4 independent accumulators break the v_add dependency chain so adds
        // co-execute with the quarter-rate v_exp_f32 TRANS ops.
        #pragma unroll
        for (int k = 0; k < 16; k += 4) {
            a0 += __builtin_amdgcn_exp2f(KS * buf[(k + 0) * 32 + lane]);
            a1 += __builtin_amdgcn_exp2f(KS * buf[(k + 1) * 32 + lane]);
            a2 += __builtin_amdgcn_exp2f(KS * buf[(k + 2) * 32 + lane]);
            a3 += __builtin_amdgcn_exp2f(KS * buf[(k + 3) * 32 + lane]);
        }
    }

    // tail (~1.1K elements), plain strided NT global loads
    for (int i = tail_start + tid; i < C_COLS; i += THREADS)
        a1 += __builtin_amdgcn_exp2f(KS * ntload(&score[row_elem + i]));

    // deterministic block tree reduction
    red[tid] = (a0 + a1) + (a2 + a3);
    __syncthreads();
    #pragma unroll
    for (int off = THREADS / 2; off > 0; off >>= 1) {
        if (tid < off) red[tid] += red[tid + off];
        __syncthreads();
    }

    if (tid == 0) {
        const float total  = red[0];                      // sum_j exp(S*score)
        const int   lab    = labels[row];                 // 0 or 1
        const float margin = lab ? 0.4f : 0.1f;
        const float target = ntload(&score[row_elem + lab]);
        const float numer  = S_SCALE * (target - margin);
        const float excl   = total - __builtin_amdgcn_exp2f(KS * target);
        const float denom  = __builtin_amdgcn_exp2f(numer * LOG2E) + excl;
        rowL[row] = numer - __builtin_amdgcn_logf(denom) * LN2;
    }
}

__global__ __launch_bounds__(THREADS)
void amsoftmax_reduce_kernel(const float* __restrict__ rowL,
                             float* __restrict__ out) {
    __shared__ float red[THREADS];
    float acc = 0.0f;
    for (int i = threadIdx.x; i < N_ROWS; i += THREADS) acc += rowL[i];
    red[threadIdx.x] = acc;
    __syncthreads();
    #pragma unroll
    for (int off = THREADS / 2; off > 0; off >>= 1) {
        if (threadIdx.x < off) red[threadIdx.x] += red[threadIdx.x + off];
        __syncthreads();
    }
    if (threadIdx.x == 0) out[0] = -red[0] / (float)N_ROWS;
}

extern "C" void kernel_launch(void* const* d_in, const int* in_sizes, int n_in,
                              void* d_out, int out_size, void* d_ws, size_t ws_size,
                              hipStream_t stream) {
    (void)in_sizes; (void)n_in; (void)out_size; (void)ws_size;
    const float* score  = (const float*)d_in[0];
    const int*   labels = (const int*)d_in[1];
    float*       out    = (float*)d_out;
    float*       rowL   = (float*)d_ws;   // 2048 floats of scratch

    amsoftmax_row_kernel<<<N_ROWS, THREADS, 0, stream>>>(score, labels, rowL);
    amsoftmax_reduce_kernel<<<1, THREADS, 0, stream>>>(rowL, out);
}